// Chamfer_Loss_11596411699393
// MI455X (gfx1250) — compile-verified
//
#include <hip/hip_runtime.h>
#include <hip/hip_bf16.h>

typedef __attribute__((ext_vector_type(2))) float v2f;
typedef __attribute__((ext_vector_type(8))) float v8f;

#define B_ 4
#define N_ 8192
#define BN_ (B_ * N_)
#define NT_ 512                   // 8192 / 16 tiles along the "search" dimension
#define COL_CHUNKS 4
#define TPC (NT_ / COL_CHUNKS)    // 128 tiles per chunk

// v_min_num_f32 without the minnum canonicalization the compiler inserts
// around llvm.minnum.
__device__ __forceinline__ float fminq(float a, float b) {
    float r;
    asm("v_min_num_f32 %0, %1, %2" : "=v"(r) : "v"(a), "v"(b));
    return r;
}

// 3-input min: folds two WMMA outputs into the running min in one VALU op.
__device__ __forceinline__ float fmin3q(float a, float b, float c) {
    float r;
    asm("v_min3_num_f32 %0, %1, %2, %3" : "=v"(r) : "v"(a), "v"(b), "v"(c));
    return r;
}

// ---------------------------------------------------------------------------
// Init row-min buffers to +inf (uint bits; d^2 >= 0 so uint order == float order)
// ---------------------------------------------------------------------------
__global__ void init_min_kernel(unsigned int* __restrict__ minbuf, int n) {
    int i = blockIdx.x * blockDim.x + threadIdx.x;
    if (i < n) minbuf[i] = 0x7F800000u;   // +inf
}

// ---------------------------------------------------------------------------
// Pre-pack the B-side operand in the exact WMMA B-fragment lane layout:
//   out[b][tile][lane] = float2 held by `lane` for that 16-column tile.
//   lane 0-15  (hi=0): (-2*y0, -2*y1)   -> K0, K1
//   lane 16-31 (hi=1): (-2*y2, |y|^2)   -> K2, K3
// ---------------------------------------------------------------------------
__global__ void pack_bfrag_kernel(const float* __restrict__ pts,
                                  float2* __restrict__ out) {
    int i    = blockIdx.x * blockDim.x + threadIdx.x;   // grid == B_*NT_*32
    int lane = i & 31;
    int tile = (i >> 5) & (NT_ - 1);
    int b    = i >> 14;                                  // 5 + 9 bits

    int n16 = lane & 15;
    int hi  = lane >> 4;

    const float* p = pts + ((size_t)b * N_ + tile * 16 + n16) * 3;
    float x = p[0], y = p[1], z = p[2];
    float w = x * x + y * y + z * z;

    float2 r;
    r.x = hi ? -2.f * z : -2.f * x;
    r.y = hi ? w        : -2.f * y;
    out[i] = r;
}

// ---------------------------------------------------------------------------
// One wave owns TWO 16-row tiles of X and a 128-tile chunk of Y columns.
// Software-pipelined, unrolled x4: each iteration prefetches the NEXT group
// of 4 B fragments (one load clause) and computes on the previous group:
//   4 x global_load_b64 + 8 x V_WMMA_F32_16X16X4_F32 + 32 x v_min3_num_f32.
//
// d^2 = |x|^2 + |y|^2 - 2 x.y realized as:
//   A row = [x0, x1, x2, 1], B col = [-2y0, -2y1, -2y2, |y|^2],
//   C pre-loaded with |x_m|^2 broadcast across columns.
// ---------------------------------------------------------------------------
__global__ void __launch_bounds__(256)
chamfer_min_kernel(const float*  __restrict__ X,      // raw [B, N, 3] (A side)
                   const float2* __restrict__ Bfrag,  // packed [B, NT_, 32]
                   unsigned int* __restrict__ rowmin) // [B, N] uint-min accum
{
    const int lane    = threadIdx.x & 31;
    const int wave    = blockIdx.x * (blockDim.x >> 5) + (threadIdx.x >> 5);
    const int chunk   = wave & (COL_CHUNKS - 1);
    const int rowPair = (wave >> 2) & (NT_ / 2 - 1);
    const int b       = wave >> 10;

    const int n16 = lane & 15;
    const int hi  = lane >> 4;

    const int rt0 = rowPair * 2;
    const int rt1 = rowPair * 2 + 1;

    // ---- A fragments + C init for both row tiles (once per wave) ----
    v2f a0, a1;
    v8f c0, c1;
    {
        const float* xp = X + ((size_t)b * N_ + rt0 * 16 + n16) * 3;
        float x0 = xp[0], x1 = xp[1], x2 = xp[2];
        float xn = x0 * x0 + x1 * x1 + x2 * x2;
        a0.x = hi ? x2 : x0;
        a0.y = hi ? 1.0f : x1;
#pragma unroll
        for (int v = 0; v < 8; ++v) c0[v] = __shfl(xn, v + 8 * hi, 32);
    }
    {
        const float* xp = X + ((size_t)b * N_ + rt1 * 16 + n16) * 3;
        float x0 = xp[0], x1 = xp[1], x2 = xp[2];
        float xn = x0 * x0 + x1 * x1 + x2 * x2;
        a1.x = hi ? x2 : x0;
        a1.y = hi ? 1.0f : x1;
#pragma unroll
        for (int v = 0; v < 8; ++v) c1[v] = __shfl(xn, v + 8 * hi, 32);
    }

    v8f run0, run1;
#pragma unroll
    for (int v = 0; v < 8; ++v) { run0[v] = __builtin_inff(); run1[v] = __builtin_inff(); }

    const float2* bb = Bfrag + ((size_t)b * NT_ + chunk * TPC) * 32 + lane;

    // ---- software pipeline: preload group 0 ----
    float2 y0 = bb[0 * 32];
    float2 y1 = bb[1 * 32];
    float2 y2 = bb[2 * 32];
    float2 y3 = bb[3 * 32];

    for (int t = 0; t < TPC; t += 4) {
        // Prefetch the NEXT group (clamped on the last iteration; `t` is
        // wave-uniform so this is a scalar select, no divergence, no OOB).
        const int tn = (t + 4 < TPC) ? (t + 4) : t;
        float2 p0 = bb[(tn + 0) * 32];
        float2 p1 = bb[(tn + 1) * 32];
        float2 p2 = bb[(tn + 2) * 32];
        float2 p3 = bb[(tn + 3) * 32];

        // Compute on the PREVIOUS group (already in registers).
        v2f bf0; bf0.x = y0.x; bf0.y = y0.y;
        v2f bf1; bf1.x = y1.x; bf1.y = y1.y;
        v2f bf2; bf2.x = y2.x; bf2.y = y2.y;
        v2f bf3; bf3.x = y3.x; bf3.y = y3.y;

        v8f d00 = __builtin_amdgcn_wmma_f32_16x16x4_f32(false, a0, false, bf0, (short)0, c0, false, false);
        v8f d01 = __builtin_amdgcn_wmma_f32_16x16x4_f32(false, a1, false, bf0, (short)0, c1, false, false);
        v8f d10 = __builtin_amdgcn_wmma_f32_16x16x4_f32(false, a0, false, bf1, (short)0, c0, false, false);
        v8f d11 = __builtin_amdgcn_wmma_f32_16x16x4_f32(false, a1, false, bf1, (short)0, c1, false, false);

#pragma unroll
        for (int v = 0; v < 8; ++v) run0[v] = fmin3q(run0[v], d00[v], d10[v]);
#pragma unroll
        for (int v = 0; v < 8; ++v) run1[v] = fmin3q(run1[v], d01[v], d11[v]);

        v8f d20 = __builtin_amdgcn_wmma_f32_16x16x4_f32(false, a0, false, bf2, (short)0, c0, false, false);
        v8f d21 = __builtin_amdgcn_wmma_f32_16x16x4_f32(false, a1, false, bf2, (short)0, c1, false, false);
        v8f d30 = __builtin_amdgcn_wmma_f32_16x16x4_f32(false, a0, false, bf3, (short)0, c0, false, false);
        v8f d31 = __builtin_amdgcn_wmma_f32_16x16x4_f32(false, a1, false, bf3, (short)0, c1, false, false);

#pragma unroll
        for (int v = 0; v < 8; ++v) run0[v] = fmin3q(run0[v], d20[v], d30[v]);
#pragma unroll
        for (int v = 0; v < 8; ++v) run1[v] = fmin3q(run1[v], d21[v], d31[v]);

        // Rotate pipeline registers.
        y0 = p0; y1 = p1; y2 = p2; y3 = p3;
    }

    // ---- cross-lane min over the 16 columns within each half-wave ----
#pragma unroll
    for (int off = 1; off < 16; off <<= 1) {
#pragma unroll
        for (int v = 0; v < 8; ++v) {
            run0[v] = fminq(run0[v], __shfl_xor(run0[v], off, 32));
            run1[v] = fminq(run1[v], __shfl_xor(run1[v], off, 32));
        }
    }

    // ---- combine across column chunks (uint atomicMin on non-negative floats) ----
    if (n16 == 0) {
#pragma unroll
        for (int v = 0; v < 8; ++v) {
            float m0 = fmaxf(run0[v], 0.0f);   // clamp commutes with min
            float m1 = fmaxf(run1[v], 0.0f);
            atomicMin(&rowmin[(size_t)b * N_ + rt0 * 16 + v + 8 * hi],
                      __float_as_uint(m0));
            atomicMin(&rowmin[(size_t)b * N_ + rt1 * 16 + v + 8 * hi],
                      __float_as_uint(m1));
        }
    }
}

// ---------------------------------------------------------------------------
// loss = mean(cham_x) + mean(cham_y) = (sum of all 2*BN mins) / BN
// ---------------------------------------------------------------------------
__global__ void reduce_kernel(const unsigned int* __restrict__ minbuf,
                              float* __restrict__ out) {
    __shared__ float sdata[256];
    float s = 0.f;
    for (int i = threadIdx.x; i < 2 * BN_; i += 256)
        s += __uint_as_float(minbuf[i]);
    sdata[threadIdx.x] = s;
    __syncthreads();
    for (int stride = 128; stride > 0; stride >>= 1) {
        if (threadIdx.x < stride) sdata[threadIdx.x] += sdata[threadIdx.x + stride];
        __syncthreads();
    }
    if (threadIdx.x == 0) out[0] = sdata[0] / (float)BN_;
}

// ---------------------------------------------------------------------------
extern "C" void kernel_launch(void* const* d_in, const int* in_sizes, int n_in,
                              void* d_out, int out_size, void* d_ws, size_t ws_size,
                              hipStream_t stream) {
    (void)in_sizes; (void)n_in; (void)out_size; (void)ws_size;

    const float* pred = (const float*)d_in[0];   // [4, 8192, 3] f32
    const float* targ = (const float*)d_in[1];   // [4, 8192, 3] f32
    float*       out  = (float*)d_out;           // scalar f32

    // Workspace layout:
    //   [0, 2*BN*4)                      : uint min buffers (cham_x | cham_y)
    //   then B_*NT_*32 float2 pred frags : 512 KB
    //   then B_*NT_*32 float2 targ frags : 512 KB
    unsigned int* minbuf  = (unsigned int*)d_ws;
    float2*       bf_pred = (float2*)((char*)d_ws + (size_t)2 * BN_ * sizeof(unsigned int));
    float2*       bf_targ = bf_pred + (size_t)B_ * NT_ * 32;

    init_min_kernel<<<(2 * BN_ + 255) / 256, 256, 0, stream>>>(minbuf, 2 * BN_);
    pack_bfrag_kernel<<<(B_ * NT_ * 32) / 256, 256, 0, stream>>>(pred, bf_pred);
    pack_bfrag_kernel<<<(B_ * NT_ * 32) / 256, 256, 0, stream>>>(targ, bf_targ);

    const int waves  = B_ * (NT_ / 2) * COL_CHUNKS;   // 4096 waves per pass
    const int blocks = waves / 8;                     // 8 waves (256 thr) per block

    // cham_x: for each pred point, min dist to targets
    chamfer_min_kernel<<<blocks, 256, 0, stream>>>(pred, bf_targ, minbuf);
    // cham_y: swap roles
    chamfer_min_kernel<<<blocks, 256, 0, stream>>>(targ, bf_pred, minbuf + BN_);

    reduce_kernel<<<1, 256, 0, stream>>>(minbuf, out);
}